// SemanticAttention_37134287241397
// MI455X (gfx1250) — compile-verified
//
#include <hip/hip_runtime.h>
#include <math.h>

// Problem constants (from reference)
#define NN    8192          // N
#define MG    3             // M graphs
#define EE    262144        // E edges per graph (2^18)
#define DIN   256           // IN_DIM
#define HID   128           // hidden
#define ROWS  (NN*MG)       // 24576 rows of the (N*M, 256) GEMM

typedef __attribute__((ext_vector_type(2))) float v2f;
typedef __attribute__((ext_vector_type(8))) float v8f;

// ---------------------------------------------------------------------------
// Kernel 0: transpose W1 (256x128 -> W1T 128x256) and fold bias c = b1 . W2
// ---------------------------------------------------------------------------
__global__ __launch_bounds__(256) void k_prep(const float* __restrict__ W1,
                                              const float* __restrict__ b1,
                                              const float* __restrict__ W2,
                                              float* __restrict__ w1t,
                                              float* __restrict__ cout) {
  __shared__ float cbuf[HID];
  const int tid = threadIdx.x;
  for (int idx = tid; idx < DIN * HID; idx += 256) {
    int k = idx >> 7;        // row of W1 (0..255)
    int h = idx & 127;       // col of W1 (0..127)
    w1t[h * DIN + k] = W1[idx];
  }
  if (tid < HID) cbuf[tid] = b1[tid] * W2[tid];
  __syncthreads();
  if (tid == 0) {
    float s = 0.f;
    for (int i = 0; i < HID; ++i) s += cbuf[i];   // fixed order: deterministic
    *cout = s;
  }
}

// ---------------------------------------------------------------------------
// Kernel 1: H = z2d @ W1 via V_WMMA_F32_16X16X4_F32, fused (.+b1)@W2,
// leaky_relu, deterministic per-block partial sums per graph m.
// Block = 256 threads = 8 waves; each wave computes one 16-row tile.
// ---------------------------------------------------------------------------
__global__ __launch_bounds__(256) void k_gemm_reduce(const float* __restrict__ z,
                                                     const float* __restrict__ w1t,
                                                     const float* __restrict__ W2,
                                                     const float* __restrict__ cptr,
                                                     float* __restrict__ partials) {
  __shared__ float wbuf[128];                 // 8 waves * 16 rows
  const int tid  = threadIdx.x;
  const int wave = tid >> 5;
  const int lane = tid & 31;
  const int half = lane >> 4;                 // 0: lanes 0-15, 1: lanes 16-31
  const int l16  = lane & 15;
  const int row0 = blockIdx.x * 128 + wave * 16;

  v8f acc[8];                                 // 8 column tiles of 16 -> HID=128
  #pragma unroll
  for (int i = 0; i < 8; ++i)
    #pragma unroll
    for (int j = 0; j < 8; ++j) acc[i][j] = 0.f;

  const float* zrow = z + (size_t)(row0 + l16) * DIN;   // A rows (both halves)

  for (int kk = 0; kk < DIN / 4; ++kk) {      // 64 K-steps of 4
    const int kb = 4 * kk + 2 * half;         // K, K+1 in the two A/B VGPRs
    // A fragment: z[row0+l16][kb..kb+1]
    const v2f a = *reinterpret_cast<const v2f*>(zrow + kb);
    #pragma unroll
    for (int nt = 0; nt < 8; ++nt) {
      // B fragment: W1[kb..kb+1][16*nt + l16]  == W1T[col][kb..kb+1] (contig)
      const v2f b = *reinterpret_cast<const v2f*>(w1t + (size_t)(16 * nt + l16) * DIN + kb);
      acc[nt] = __builtin_amdgcn_wmma_f32_16x16x4_f32(
          false, a, false, b, (short)0, acc[nt], false, false);
    }
  }

  // Epilogue: hv[row] = sum_cols (H[row][col]) * W2[col]; then + (b1.W2)
  float w2v[8];
  #pragma unroll
  for (int nt = 0; nt < 8; ++nt) w2v[nt] = W2[16 * nt + l16];
  const float cval = *cptr;

  #pragma unroll
  for (int j = 0; j < 8; ++j) {               // row (j + 8*half) of the tile
    float s = 0.f;
    #pragma unroll
    for (int nt = 0; nt < 8; ++nt) s += acc[nt][j] * w2v[nt];
    // reduce over the 16 lanes of this half (xor 1,2,4,8 stays in the half)
    #pragma unroll
    for (int m = 1; m < 16; m <<= 1) s += __shfl_xor(s, m, 32);
    if (l16 == 0) {
      float w  = s + cval;
      float lk = (w > 0.f) ? w : 0.01f * w;   // leaky_relu, slope 0.01
      wbuf[wave * 16 + j + 8 * half] = lk;
    }
  }
  __syncthreads();
  if (tid == 0) {                             // deterministic per-block sums
    float ps[3] = {0.f, 0.f, 0.f};
    const int base = blockIdx.x * 128;        // global row = n*3 + m
    for (int r = 0; r < 128; ++r) ps[(base + r) % 3] += wbuf[r];
    partials[blockIdx.x * 3 + 0] = ps[0];
    partials[blockIdx.x * 3 + 1] = ps[1];
    partials[blockIdx.x * 3 + 2] = ps[2];
  }
}

// ---------------------------------------------------------------------------
// Kernel 2: deterministic final reduce + mean over N + softmax over M=3
// ---------------------------------------------------------------------------
__global__ void k_beta(const float* __restrict__ partials, float* __restrict__ beta) {
  if (threadIdx.x == 0 && blockIdx.x == 0) {
    float wm[3];
    for (int m = 0; m < 3; ++m) {
      float s = 0.f;
      for (int b = 0; b < 192; ++b) s += partials[b * 3 + m];
      wm[m] = s / (float)NN;
    }
    float mx = fmaxf(wm[0], fmaxf(wm[1], wm[2]));
    float e0 = expf(wm[0] - mx), e1 = expf(wm[1] - mx), e2 = expf(wm[2] - mx);
    float inv = 1.f / (e0 + e1 + e2);
    beta[0] = e0 * inv; beta[1] = e1 * inv; beta[2] = e2 * inv;
  }
}

// ---------------------------------------------------------------------------
// Kernel 3: zero-fill atten (N*N floats) with float4 stores
// ---------------------------------------------------------------------------
__global__ __launch_bounds__(256) void k_zero(float4* __restrict__ out4) {
  size_t idx = (size_t)blockIdx.x * 256 + threadIdx.x;   // 16,777,216 total
  out4[idx] = make_float4(0.f, 0.f, 0.f, 0.f);
}

// ---------------------------------------------------------------------------
// Kernel 4: z_out[n,d] = sum_m beta[m] * z[n,m,d]   (float4 vectorized)
// ---------------------------------------------------------------------------
__global__ __launch_bounds__(256) void k_zout(const float* __restrict__ z,
                                              const float* __restrict__ beta,
                                              float4* __restrict__ out4) {
  const int idx = blockIdx.x * 256 + threadIdx.x;        // 524,288 total
  const float b0 = beta[0], b1 = beta[1], b2 = beta[2];
  const int n = idx >> 6;          // 64 float4 per row of 256
  const int q = idx & 63;
  const float4* z4 = reinterpret_cast<const float4*>(z);
  const size_t base = (size_t)n * 3 * 64;
  float4 a = z4[base + q];
  float4 b = z4[base + 64 + q];
  float4 c = z4[base + 128 + q];
  float4 r;
  r.x = b0 * a.x + b1 * b.x + b2 * c.x;
  r.y = b0 * a.y + b1 * b.y + b2 * c.y;
  r.z = b0 * a.z + b1 * b.z + b2 * c.z;
  r.w = b0 * a.w + b1 * b.w + b2 * c.w;
  out4[idx] = r;
}

// ---------------------------------------------------------------------------
// Kernel 5: fused scatter-add: atten[src,dst] += alpha[m,e] * beta[m]
// ---------------------------------------------------------------------------
__global__ __launch_bounds__(256) void k_scatter(const float* __restrict__ alpha,
                                                 const int* __restrict__ src,
                                                 const int* __restrict__ dst,
                                                 const float* __restrict__ beta,
                                                 float* __restrict__ atten) {
  const int gid = blockIdx.x * 256 + threadIdx.x;        // < 786,432
  const int m = gid >> 18;                               // E = 2^18
  const float v = alpha[gid] * beta[m];
  const int s = src[gid];
  const int d = dst[gid];
  unsafeAtomicAdd(&atten[(size_t)s * NN + d], v);        // global_atomic_add_f32
}

// ---------------------------------------------------------------------------
extern "C" void kernel_launch(void* const* d_in, const int* in_sizes, int n_in,
                              void* d_out, int out_size, void* d_ws, size_t ws_size,
                              hipStream_t stream) {
  const float* z        = (const float*)d_in[0];   // (N, M, 256)
  const float* alpha    = (const float*)d_in[1];   // (M, E)
  const float* W1       = (const float*)d_in[2];   // (256, 128)
  const float* b1       = (const float*)d_in[3];   // (128,)
  const float* W2       = (const float*)d_in[4];   // (128, 1)
  const int*   edge_src = (const int*)d_in[5];     // (M, E)
  const int*   edge_dst = (const int*)d_in[6];     // (M, E)

  float* out   = (float*)d_out;
  float* zout  = out;                              // N*256 floats
  float* atten = out + (size_t)NN * DIN;           // N*N floats

  // workspace layout (floats)
  float* ws       = (float*)d_ws;
  float* w1t      = ws;                            // 128*256 = 32768
  float* cscal    = ws + 32768;                    // 1
  float* partials = ws + 32769;                    // 192*3 = 576
  float* beta     = ws + 32769 + 576;              // 3

  k_prep<<<1, 256, 0, stream>>>(W1, b1, W2, w1t, cscal);
  k_gemm_reduce<<<ROWS / 128, 256, 0, stream>>>(z, w1t, W2, cscal, partials);
  k_beta<<<1, 32, 0, stream>>>(partials, beta);
  k_zero<<<(NN * (size_t)NN / 4) / 256, 256, 0, stream>>>((float4*)atten);
  k_zout<<<(NN * DIN / 4) / 256, 256, 0, stream>>>(z, beta, (float4*)zout);
  k_scatter<<<(MG * EE) / 256, 256, 0, stream>>>(alpha, edge_src, edge_dst, beta, atten);
}